// Mnist_Classifier_40269613367562
// MI455X (gfx1250) — compile-verified
//
#include <hip/hip_runtime.h>
#include <math.h>

#define DINL __device__ __forceinline__

typedef __attribute__((ext_vector_type(2))) float v2f;
typedef __attribute__((ext_vector_type(8))) float v8f;

#define PI_D 3.14159265358979323846

// ---------------- problem sizes ----------------
// BATCH=32, B_IN=30, B_L1=10, B_OUT=5, K_IN=1, K_L1=100, K_L2=200, K_OUT=10
#define NB   32
#define K1C  100
#define K2C  200
#define NA1  60     // 2*B_IN
#define NA2  20     // 2*B_L1
#define NA3  10     // 2*B_OUT

#define SC1F 0.03872983346207417f   // 1/sqrt(2000/3)
#define SC2F 0.06324555320336759f   // 1/sqrt(250)

__constant__ int P1c[10] = {0,1,10,35,84,165,286,455,684,969}; // prefix of (2l+1)^2, l<10 (total 1330)
__constant__ int P2c[5]  = {0,1,10,35,84};                     // l<5 (total 165)

// ---------------- helpers ----------------
DINL int iabs_(int v){ return v < 0 ? -v : v; }
DINL int imax_(int a, int b){ return a > b ? a : b; }

DINL double factd(int k){ double r = 1.0; for (int i = 2; i <= k; ++i) r *= (double)i; return r; }
DINL double ipowd(double x, int e){ double r = 1.0; for (int i = 0; i < e; ++i) r *= x; return r; }

// Wigner small-d, d^l_{m,n}(beta), double precision (l<=9 -> fact(18) fits in f64)
DINL double wig_d(int l, int m, int n, double beta){
  double cb = cos(beta * 0.5), sb = sin(beta * 0.5);
  double pref = sqrt(factd(l+m) * factd(l-m) * factd(l+n) * factd(l-n));
  int s0 = imax_(0, n - m);
  int s1 = (l + n < l - m) ? (l + n) : (l - m);
  double acc = 0.0;
  for (int s = s0; s <= s1; ++s){
    double den = factd(l+n-s) * factd(s) * factd(m-n+s) * factd(l-m-s);
    double sgn = ((m - n + s) & 1) ? -1.0 : 1.0;
    acc += sgn / den * ipowd(cb, 2*l + n - m - 2*s) * ipowd(sb, m - n + 2*s);
  }
  return pref * acc;
}

DINL double quadw(int bq, int j){
  double beta = PI_D * (2*j + 1) / (4.0 * bq);
  double s = 0.0;
  for (int k = 1; k < 2*bq; k += 2) s += sin(k * beta) / (double)k;
  return 2.0 / bq * sin(beta) * s;
}

DINL void dec_lm(int lm, int& l, int& m){
  l = 0; while ((l+1)*(l+1) <= lm) ++l;
  m = lm - l*l - l;
}
DINL void dec_pair(int pair, int& l, int& m, int& n){
  l = 0; int off = pair;
  while (off >= (2*l+1)*(2*l+1)){ off -= (2*l+1)*(2*l+1); ++l; }
  int tl = 2*l + 1;
  m = off / tl - l;
  n = off % tl - l;
}

// ---------------- constant-table generation (runs every call; deterministic) ----------------
__global__ void k_weights(float* wIn, float* wL1, float* wO){
  int t = blockIdx.x * blockDim.x + threadIdx.x;
  if      (t < 60) wIn[t]      = (float)quadw(30, t);
  else if (t < 80) wL1[t - 60] = (float)quadw(10, t - 60);
  else if (t < 90) wO[t - 80]  = (float)quadw(5,  t - 80);
}

__global__ void k_tab_d1(const float* __restrict__ wIn, float* __restrict__ tD1){
  int idx = blockIdx.x * blockDim.x + threadIdx.x;
  if (idx >= 100*60) return;
  int lm = idx / 60, z = idx % 60, l, m;
  dec_lm(lm, l, m);
  double beta = PI_D * (2*z + 1) / 120.0;
  tD1[idx] = (float)(wig_d(l, m, 0, beta)) * wIn[z];
}

__global__ void k_tab_b1(float* __restrict__ br, float* __restrict__ bi){
  int idx = blockIdx.x * blockDim.x + threadIdx.x;
  if (idx >= 100*60) return;
  int lm = idx / 60, p = idx % 60, l, m;
  dec_lm(lm, l, m);
  double dh = wig_d(l, m, 0, PI_D * 0.5);
  double ang = -2.0 * PI_D * m * p / 60.0;
  br[idx] = (float)(cos(ang) * dh);
  bi[idx] = (float)(sin(ang) * dh);
}

__global__ void k_tab_dinv1(float* __restrict__ t){
  int idx = blockIdx.x * blockDim.x + threadIdx.x;
  if (idx >= 1330*20) return;
  int pair = idx / 20, z = idx % 20, l, m, n;
  dec_pair(pair, l, m, n);
  double beta = PI_D * (2*z + 1) / 40.0;
  t[idx] = (float)((2*l + 1) * wig_d(l, m, n, beta));
}

__global__ void k_tab_d2(const float* __restrict__ wL1, float* __restrict__ t){
  int idx = blockIdx.x * blockDim.x + threadIdx.x;
  if (idx >= 165*20) return;
  int pair = idx / 20, z = idx % 20, l, m, n;
  dec_pair(pair, l, m, n);
  double beta = PI_D * (2*z + 1) / 40.0;
  t[idx] = (float)(wig_d(l, m, n, beta)) * wL1[z];
}

__global__ void k_tab_dinv2(float* __restrict__ t){
  int idx = blockIdx.x * blockDim.x + threadIdx.x;
  if (idx >= 165*10) return;
  int pair = idx / 10, z = idx % 10, l, m, n;
  dec_pair(pair, l, m, n);
  double beta = PI_D * (2*z + 1) / 20.0;
  t[idx] = (float)((2*l + 1) * wig_d(l, m, n, beta));
}

__global__ void k_tab_b2(float* __restrict__ br, float* __restrict__ bi){
  int idx = blockIdx.x * blockDim.x + threadIdx.x;
  if (idx >= 165*20) return;
  int pair = idx / 20, p = idx % 20, l, m, n;
  dec_pair(pair, l, m, n);
  double dh = wig_d(l, m, n, PI_D * 0.5);
  double ang = -2.0 * PI_D * m * p / 20.0;
  br[idx] = (float)(cos(ang) * dh);
  bi[idx] = (float)(sin(ang) * dh);
}

// ---------------- stage 1: S2 analysis ----------------
// xhat1[lm=(l,m)][b] = sum_z D1[lm][z] * sum_t x[b,0,z,t] e^{-2pi i m t/60}
__global__ void k_xhat1(const float* __restrict__ x, const float* __restrict__ tD1,
                        float* __restrict__ xr, float* __restrict__ xi){
  int idx = blockIdx.x * blockDim.x + threadIdx.x;
  if (idx >= 100*NB) return;
  int lm = idx >> 5, b = idx & 31, l, m;
  dec_lm(lm, l, m);
  const float* xb = x + (size_t)b * 3600;
  float dc = cosf(-(float)PI_D * m / 30.0f);
  float dsn = sinf(-(float)PI_D * m / 30.0f);
  float accr = 0.f, acci = 0.f;
  for (int z = 0; z < 60; ++z){
    float d = tD1[lm*60 + z];
    const float* row = xb + z*60;
    float cr = 1.f, ci = 0.f, sr = 0.f, si = 0.f;
    for (int tt = 0; tt < 60; ++tt){
      float v = row[tt];
      sr += v * cr; si += v * ci;
      float t2 = cr*dc - ci*dsn;
      ci = cr*dsn + ci*dc;
      cr = t2;
    }
    accr += d * sr; acci += d * si;
  }
  xr[idx] = accr; xi[idx] = acci;   // layout [lm][b]
}

// yhat1[lm][o] = SC1 * sum_p k1[0,o,p] * BASIS1[lm][p]
__global__ void k_yhat1(const float* __restrict__ k1,
                        const float* __restrict__ b1r, const float* __restrict__ b1i,
                        float* __restrict__ yr, float* __restrict__ yi){
  int idx = blockIdx.x * blockDim.x + threadIdx.x;
  if (idx >= 100*K1C) return;
  int lm = idx / K1C, o = idx % K1C;
  float ar = 0.f, ai = 0.f;
  for (int p = 0; p < 60; ++p){
    float kv = k1[o*60 + p] * SC1F;
    ar += kv * b1r[lm*60 + p];
    ai += kv * b1i[lm*60 + p];
  }
  yr[idx] = ar; yi[idx] = ai;       // layout [lm][o]
}

// Fused: build freq tile G[b,o,z] (20x20) in LDS from xhat1*conj(yhat1)*DINV1,
// inverse-2D-DFT in LDS, +b1, ReLU, store real h1. No 205MB complex buffer.
__global__ __launch_bounds__(256) void k_s2_synth(
    const float* __restrict__ xr, const float* __restrict__ xi,
    const float* __restrict__ yr, const float* __restrict__ yi,
    const float* __restrict__ dinv1, const float* __restrict__ b1,
    float* __restrict__ h1){
  int b = blockIdx.x / 100, o = blockIdx.x % 100;
  __shared__ float sxr[100], sxi[100], syr[100], syi[100];
  __shared__ float gr[400], gi[400], jr[400], ji[400];
  __shared__ float c20[20], s20[20];
  for (int k = threadIdx.x; k < 100; k += 256){
    sxr[k] = xr[k*32 + b]; sxi[k] = xi[k*32 + b];
    syr[k] = yr[k*100 + o]; syi[k] = yi[k*100 + o];
  }
  for (int k = threadIdx.x; k < 20; k += 256){
    float a = 2.f * (float)PI_D * k / 20.f;
    c20[k] = cosf(a); s20[k] = sinf(a);
  }
  __syncthreads();
  for (int z = 0; z < 20; ++z){
    for (int q = threadIdx.x; q < 400; q += 256){
      int mu = q / 20, nu = q % 20;
      int m = (mu <= 9) ? mu : mu - 20;
      int n = (nu <= 9) ? nu : nu - 20;
      int am = iabs_(m), an = iabs_(n);
      float accr = 0.f, acci = 0.f;
      if (am <= 9 && an <= 9){
        int lmin = imax_(am, an);
        for (int l = lmin; l < 10; ++l){
          int tl = 2*l + 1;
          int p = P1c[l] + (m + l)*tl + (n + l);
          float d = dinv1[p*20 + z];
          float xrv = sxr[l*l + m + l], xiv = sxi[l*l + m + l];
          float yrv = syr[l*l + n + l], yiv = syi[l*l + n + l];
          accr += d * (xrv*yrv + xiv*yiv);   // x * conj(y)
          acci += d * (xiv*yrv - xrv*yiv);
        }
      }
      gr[q] = accr; gi[q] = acci;
    }
    __syncthreads();
    for (int q = threadIdx.x; q < 400; q += 256){   // J[mu,beta] = sum_nu G e^{+i th nu beta}
      int mu = q / 20, be = q % 20;
      float ar = 0.f, ai = 0.f;
      for (int nu = 0; nu < 20; ++nu){
        int id = (nu * be) % 20;
        float c = c20[id], s = s20[id];
        float gvr = gr[mu*20 + nu], gvi = gi[mu*20 + nu];
        ar += gvr*c - gvi*s;
        ai += gvr*s + gvi*c;
      }
      jr[q] = ar; ji[q] = ai;
    }
    __syncthreads();
    for (int q = threadIdx.x; q < 400; q += 256){   // h[al,be] = Re sum_mu J e^{+i th mu al}
      int al = q / 20, be = q % 20;
      float h = 0.f;
      for (int mu = 0; mu < 20; ++mu){
        int id = (mu * al) % 20;
        h += jr[mu*20 + be]*c20[id] - ji[mu*20 + be]*s20[id];
      }
      h += b1[o];
      h = h > 0.f ? h : 0.f;
      h1[((size_t)(b*100 + o)*20 + z)*400 + q] = h;
    }
    __syncthreads();
  }
}

// ---------------- stage 2: SO3 analysis ----------------
// Only |m|,|n|<=4 frequencies of fft2(h1) are consumed: sample 81 of 400 directly.
__global__ __launch_bounds__(256) void k_xfreq(const float* __restrict__ h1,
                                               float* __restrict__ fr, float* __restrict__ fi){
  int biz = blockIdx.x;   // (b*100+i)*20 + z
  __shared__ float sh[400];
  __shared__ float t1r[180], t1i[180];
  __shared__ float c20[20], s20[20];
  for (int k = threadIdx.x; k < 400; k += 256) sh[k] = h1[(size_t)biz*400 + k];
  for (int k = threadIdx.x; k < 20; k += 256){
    float a = 2.f * (float)PI_D * k / 20.f;
    c20[k] = cosf(a); s20[k] = sinf(a);
  }
  __syncthreads();
  for (int q = threadIdx.x; q < 180; q += 256){   // t1[m9,beta] = sum_al sh[al,be] e^{-i th m al}
    int m9 = q / 20, be = q % 20, m = m9 - 4;
    float ar = 0.f, ai = 0.f;
    for (int al = 0; al < 20; ++al){
      int id = ((m * al) % 20 + 20) % 20;
      float v = sh[al*20 + be];
      ar += v * c20[id];
      ai -= v * s20[id];
    }
    t1r[q] = ar; t1i[q] = ai;
  }
  __syncthreads();
  for (int q = threadIdx.x; q < 81; q += 256){    // xf[m9,n9] = sum_be t1 e^{-i th n be}
    int m9 = q / 9, n9 = q % 9, n = n9 - 4;
    float ar = 0.f, ai = 0.f;
    for (int be = 0; be < 20; ++be){
      int id = ((n * be) % 20 + 20) % 20;
      float c = c20[id], s = -s20[id];
      float xr = t1r[m9*20 + be], xi = t1i[m9*20 + be];
      ar += xr*c - xi*s;
      ai += xr*s + xi*c;
    }
    fr[(size_t)biz*81 + q] = ar;
    fi[(size_t)biz*81 + q] = ai;
  }
}

// xhat2[pair(m,k)][b*100+i] = sum_z xfreq[b,i,z,m,k] * D2[pair][z]
__global__ void k_xhat2(const float* __restrict__ fr, const float* __restrict__ fi,
                        const float* __restrict__ tD2,
                        float* __restrict__ xr, float* __restrict__ xi){
  int idx = blockIdx.x * blockDim.x + threadIdx.x;
  if (idx >= 165*3200) return;
  int pair = idx / 3200, r = idx % 3200;   // r = b*100+i
  int l, m, n; dec_pair(pair, l, m, n);
  int fq = (m + 4)*9 + (n + 4);
  float ar = 0.f, ai = 0.f;
  for (int z = 0; z < 20; ++z){
    float d = tD2[pair*20 + z];
    size_t q = ((size_t)r*20 + z)*81 + fq;
    ar += d * fr[q];
    ai += d * fi[q];
  }
  xr[pair*3200 + r] = ar;
  xi[pair*3200 + r] = ai;
}

// yhat2[pair(n,k)][i*208+o] = SC2 * sum_p k2[i,o,p]*BASIS2[pair][p]; N padded to 208
__global__ void k_yhat2(const float* __restrict__ k2,
                        const float* __restrict__ b2r, const float* __restrict__ b2i,
                        float* __restrict__ yr, float* __restrict__ yi){
  int idx = blockIdx.x * blockDim.x + threadIdx.x;
  if (idx >= 165*100*208) return;
  int pair = idx / 20800, rem = idx % 20800;
  int i = rem / 208, o = rem % 208;
  if (o >= 200){ yr[idx] = 0.f; yi[idx] = 0.f; return; }
  float ar = 0.f, ai = 0.f;
  for (int p = 0; p < 20; ++p){
    float kv = k2[((size_t)i*200 + o)*20 + p] * SC2F;
    ar += kv * b2r[pair*20 + p];
    ai += kv * b2i[pair*20 + p];
  }
  yr[idx] = ar; yi[idx] = ai;
}

// ---------------- the WMMA GEMM: z[m,n,b,o] = sum_{k,i} xhat2[m,k,b,i]*conj(yhat2[n,k,i,o]) ----------------
#if __has_builtin(__builtin_amdgcn_wmma_f32_16x16x4_f32)
#define HAVE_WMMA_F32X4 1
#else
#define HAVE_WMMA_F32X4 0
#endif

__global__ __launch_bounds__(832) void k_z2_gemm(
    const float* __restrict__ xr, const float* __restrict__ xi,
    const float* __restrict__ yr, const float* __restrict__ yi,
    float* __restrict__ zr, float* __restrict__ zi){
  int pair = blockIdx.x;
  int l = 0, off = pair;
  while (off >= (2*l+1)*(2*l+1)){ off -= (2*l+1)*(2*l+1); ++l; }
  const int tl  = 2*l + 1;
  const int mi  = off / tl;     // m+l
  const int ni  = off % tl;     // n+l
  const int Pl  = pair - off;   // P2[l]
  const int pA0 = Pl + mi*tl;   // pair base (m, k=-l)
  const int pB0 = Pl + ni*tl;   // pair base (n, k=-l)
  const int K   = tl * 100;

  const int wave = threadIdx.x >> 5;
  const int lane = threadIdx.x & 31;
  const int tm = wave & 1;        // 2 row-tiles (batch 32)
  const int tn = wave >> 1;       // 13 col-tiles (200 -> pad 208)
  const int half = lane >> 4;
  const int lr = lane & 15;
  const int Mrow = tm*16 + lr;
  const int Ncol = tn*16 + lr;

#if HAVE_WMMA_F32X4
  v8f aRR = {0,0,0,0,0,0,0,0};
  v8f aII = aRR, aIR = aRR, aRI = aRR;
  for (int kb = 0; kb < K; kb += 4){
    const int khi = kb / 100;
    const int i0  = kb - khi*100;          // 100 % 4 == 0: never straddles k slices
    const int ao  = (pA0 + khi)*3200 + Mrow*100 + i0 + 2*half;
    v2f ar = { xr[ao], xr[ao + 1] };
    v2f ai = { xi[ao], xi[ao + 1] };
    const int bo = (pB0 + khi)*20800 + (i0 + 2*half)*208 + Ncol;
    v2f br = { yr[bo], yr[bo + 208] };
    v2f bi = { yi[bo], yi[bo + 208] };
    __builtin_prefetch(&yr[bo + 4*208], 0, 1);
    aRR = __builtin_amdgcn_wmma_f32_16x16x4_f32(false, ar, false, br, (short)0, aRR, false, false);
    aII = __builtin_amdgcn_wmma_f32_16x16x4_f32(false, ai, false, bi, (short)0, aII, false, false);
    aIR = __builtin_amdgcn_wmma_f32_16x16x4_f32(false, ai, false, br, (short)0, aIR, false, false);
    aRI = __builtin_amdgcn_wmma_f32_16x16x4_f32(false, ar, false, bi, (short)0, aRI, false, false);
  }
  const int pMN = Pl + mi*tl + ni;
  const int o = tn*16 + lr;
  if (o < 200){
    for (int v = 0; v < 8; ++v){
      int brow = tm*16 + v + half*8;                  // C/D layout: lanes>=16 hold M=v+8
      zr[pMN*6400 + brow*200 + o] = aRR[v] + aII[v];  // Re(x*conj(y))
      zi[pMN*6400 + brow*200 + o] = aIR[v] - aRI[v];  // Im(x*conj(y))
    }
  }
#else
  float accr[8], acci[8];
  for (int v = 0; v < 8; ++v){ accr[v] = 0.f; acci[v] = 0.f; }
  const int o = tn*16 + lr;
  for (int k = 0; k < K; ++k){
    const int khi = k / 100, ii = k - khi*100;
    const float brv = (o < 200) ? yr[(pB0 + khi)*20800 + ii*208 + o] : 0.f;
    const float biv = (o < 200) ? yi[(pB0 + khi)*20800 + ii*208 + o] : 0.f;
    for (int v = 0; v < 8; ++v){
      int brow = tm*16 + v + half*8;
      float xrv = xr[(pA0 + khi)*3200 + brow*100 + ii];
      float xiv = xi[(pA0 + khi)*3200 + brow*100 + ii];
      accr[v] += xrv*brv + xiv*biv;
      acci[v] += xiv*brv - xrv*biv;
    }
  }
  const int pMN = Pl + mi*tl + ni;
  if (o < 200){
    for (int v = 0; v < 8; ++v){
      int brow = tm*16 + v + half*8;
      zr[pMN*6400 + brow*200 + o] = accr[v];
      zi[pMN*6400 + brow*200 + o] = acci[v];
    }
  }
#endif
}

// ---------------- stage 3: SO3 synthesis fused with pooling ----------------
// Per (b,o): for each z3, build 10x10 freq tile from z2*DINV2, IDFT, +b2, ReLU,
// weight by W_INT[z3], sum over (alpha,beta,z3) -> hsum[b,o]. No complex cube stored.
__global__ __launch_bounds__(128) void k_so3_synth(
    const float* __restrict__ z2r, const float* __restrict__ z2i,
    const float* __restrict__ dinv2, const float* __restrict__ b2,
    const float* __restrict__ wO, float* __restrict__ hsum){
  int b = blockIdx.x / 200, o = blockIdx.x % 200;
  __shared__ float szr[165], szi[165];
  __shared__ float gr[100], gi[100], jr2[100], ji2[100];
  __shared__ float c10[10], s10[10];
  __shared__ float red[128];
  for (int k = threadIdx.x; k < 165; k += 128){
    szr[k] = z2r[k*6400 + b*200 + o];
    szi[k] = z2i[k*6400 + b*200 + o];
  }
  for (int k = threadIdx.x; k < 10; k += 128){
    float a = 2.f * (float)PI_D * k / 10.f;
    c10[k] = cosf(a); s10[k] = sinf(a);
  }
  __syncthreads();
  float acc = 0.f;
  for (int z3 = 0; z3 < 10; ++z3){
    for (int q = threadIdx.x; q < 100; q += 128){
      int mu = q / 10, nu = q % 10;
      int m = (mu <= 4) ? mu : mu - 10;
      int n = (nu <= 4) ? nu : nu - 10;
      int am = iabs_(m), an = iabs_(n);
      float ar = 0.f, ai = 0.f;
      if (am <= 4 && an <= 4){
        int lmin = imax_(am, an);
        for (int l = lmin; l < 5; ++l){
          int tl = 2*l + 1;
          int p = P2c[l] + (m + l)*tl + (n + l);
          float d = dinv2[p*10 + z3];
          ar += d * szr[p];
          ai += d * szi[p];
        }
      }
      gr[q] = ar; gi[q] = ai;
    }
    __syncthreads();
    for (int q = threadIdx.x; q < 100; q += 128){
      int mu = q / 10, be = q % 10;
      float ar = 0.f, ai = 0.f;
      for (int nu = 0; nu < 10; ++nu){
        int id = (nu * be) % 10;
        ar += gr[mu*10 + nu]*c10[id] - gi[mu*10 + nu]*s10[id];
        ai += gr[mu*10 + nu]*s10[id] + gi[mu*10 + nu]*c10[id];
      }
      jr2[q] = ar; ji2[q] = ai;
    }
    __syncthreads();
    for (int q = threadIdx.x; q < 100; q += 128){
      int al = q / 10, be = q % 10;
      float h = 0.f;
      for (int mu = 0; mu < 10; ++mu){
        int id = (mu * al) % 10;
        h += jr2[mu*10 + be]*c10[id] - ji2[mu*10 + be]*s10[id];
      }
      h += b2[o];
      h = h > 0.f ? h : 0.f;
      acc += wO[z3] * h;
    }
    __syncthreads();
  }
  red[threadIdx.x] = acc;
  __syncthreads();
  for (int s = 64; s > 0; s >>= 1){
    if (threadIdx.x < (unsigned)s) red[threadIdx.x] += red[threadIdx.x + s];
    __syncthreads();
  }
  if (threadIdx.x == 0) hsum[b*200 + o] = red[0];
}

// ---------------- final linear ----------------
__global__ void k_final(const float* __restrict__ hsum, const float* __restrict__ W,
                        const float* __restrict__ bl, float* __restrict__ out){
  int idx = blockIdx.x * blockDim.x + threadIdx.x;
  if (idx >= NB*10) return;
  int b = idx / 10, f = idx % 10;
  float acc = bl[f];
  for (int o = 0; o < 200; ++o) acc += hsum[b*200 + o] * W[f*200 + o];
  out[idx] = acc;
}

// ---------------- host launcher ----------------
#define CDIV(a,b) (((a)+(b)-1)/(b))

extern "C" void kernel_launch(void* const* d_in, const int* in_sizes, int n_in,
                              void* d_out, int out_size, void* d_ws, size_t ws_size,
                              hipStream_t stream){
  (void)in_sizes; (void)n_in; (void)out_size; (void)ws_size;
  const float* x  = (const float*)d_in[0];
  const float* k1 = (const float*)d_in[1];
  const float* b1 = (const float*)d_in[2];
  const float* k2 = (const float*)d_in[3];
  const float* b2 = (const float*)d_in[4];
  const float* W  = (const float*)d_in[5];
  const float* bl = (const float*)d_in[6];
  float* out = (float*)d_out;

  float* ws = (float*)d_ws;
  size_t off = 0;
  auto alloc = [&](size_t n){ float* p = ws + off; off += (n + 63) & ~(size_t)63; return p; };
  float* wIn  = alloc(60);
  float* wL1  = alloc(20);
  float* wO   = alloc(10);
  float* tD1  = alloc(6000);
  float* tB1r = alloc(6000);
  float* tB1i = alloc(6000);
  float* tDI1 = alloc(26600);
  float* tD2  = alloc(3300);
  float* tDI2 = alloc(1650);
  float* tB2r = alloc(3300);
  float* tB2i = alloc(3300);
  float* xh1r = alloc(3200);
  float* xh1i = alloc(3200);
  float* yh1r = alloc(10000);
  float* yh1i = alloc(10000);
  float* H1   = alloc(25600000);  // 32*100*20*20*20 f32
  float* XFR  = alloc(5184000);   // 64000*81
  float* XFI  = alloc(5184000);
  float* XH2R = alloc(528000);    // 165*3200
  float* XH2I = alloc(528000);
  float* HSUM = alloc(6400);
  // alias late buffers inside H1 (H1 is dead once k_xfreq completes; stream-ordered)
  float* YH2R = H1;               // 165*20800 = 3,432,000
  float* YH2I = H1 + 3432000;
  float* Z2R  = H1 + 6864000;     // 165*6400 = 1,056,000
  float* Z2I  = H1 + 7920000;     // ends at 8,976,000 < 25,600,000

  k_weights  <<<1, 128, 0, stream>>>(wIn, wL1, wO);
  k_tab_d1   <<<CDIV(6000,256), 256, 0, stream>>>(wIn, tD1);
  k_tab_b1   <<<CDIV(6000,256), 256, 0, stream>>>(tB1r, tB1i);
  k_tab_dinv1<<<CDIV(26600,256),256, 0, stream>>>(tDI1);
  k_tab_d2   <<<CDIV(3300,256), 256, 0, stream>>>(wL1, tD2);
  k_tab_dinv2<<<CDIV(1650,256), 256, 0, stream>>>(tDI2);
  k_tab_b2   <<<CDIV(3300,256), 256, 0, stream>>>(tB2r, tB2i);

  k_xhat1    <<<CDIV(3200,256), 256, 0, stream>>>(x, tD1, xh1r, xh1i);
  k_yhat1    <<<CDIV(10000,256),256, 0, stream>>>(k1, tB1r, tB1i, yh1r, yh1i);
  k_s2_synth <<<3200, 256, 0, stream>>>(xh1r, xh1i, yh1r, yh1i, tDI1, b1, H1);

  k_xfreq    <<<64000, 256, 0, stream>>>(H1, XFR, XFI);
  k_yhat2    <<<CDIV(3432000,256), 256, 0, stream>>>(k2, tB2r, tB2i, YH2R, YH2I);
  k_xhat2    <<<CDIV(528000,256), 256, 0, stream>>>(XFR, XFI, tD2, XH2R, XH2I);

  k_z2_gemm  <<<165, 832, 0, stream>>>(XH2R, XH2I, YH2R, YH2I, Z2R, Z2I);

  k_so3_synth<<<6400, 128, 0, stream>>>(Z2R, Z2I, tDI2, b2, wO, HSUM);
  k_final    <<<CDIV(320,64), 64, 0, stream>>>(HSUM, W, bl, out);
}